// GCN_38044820308056
// MI455X (gfx1250) — compile-verified
//
#include <hip/hip_runtime.h>
#include <hip/hip_bf16.h>

typedef float v2f __attribute__((ext_vector_type(2)));
typedef float v8f __attribute__((ext_vector_type(8)));

#define HID 64

// ---------------- degree / normalization ----------------
__global__ void k_deg_init(float* __restrict__ deg, int N) {
    int i = blockIdx.x * blockDim.x + threadIdx.x;
    if (i < N) deg[i] = 1.0f;                       // self-loop contributes 1
}

__global__ void k_deg_count(const int* __restrict__ dst, float* __restrict__ deg, int E) {
    int e = blockIdx.x * blockDim.x + threadIdx.x;
    if (e < E) atomicAdd(&deg[dst[e]], 1.0f);
}

__global__ void k_deg_rsqrt(float* __restrict__ dinv, int N) {
    int i = blockIdx.x * blockDim.x + threadIdx.x;
    if (i < N) dinv[i] = rsqrtf(dinv[i]);           // deg >= 1 always (self-loop)
}

// ---------------- layer-1 linear: [N,9] @ [9,64] ----------------
__global__ void k_lin_in(const float* __restrict__ x, const float* __restrict__ W,
                         float* __restrict__ out, int N) {
    int tid = blockIdx.x * blockDim.x + threadIdx.x;
    if (tid >= N * HID) return;
    int i = tid >> 6, f = tid & 63;
    const float* xr = x + i * 9;
    float acc = 0.0f;
#pragma unroll
    for (int k = 0; k < 9; ++k) acc = fmaf(xr[k], W[k * HID + f], acc);
    out[tid] = acc;
}

// ---------------- WMMA GEMM: [N,64] @ [64,64] (fp32, exact) ----------------
// Block = 256 threads = 8 waves. Block covers 32 rows x 64 cols:
// wave w: M-tile = w>>2 (2 tiles of 16 rows), N-tile = w&3 (4 tiles of 16 cols).
// Weight matrix staged in LDS (16 KB).
__global__ __launch_bounds__(256) void k_gemm_wmma(const float* __restrict__ A,
                                                   const float* __restrict__ W,
                                                   float* __restrict__ Bout, int N) {
    __shared__ float w[HID * HID];
    for (int idx = threadIdx.x; idx < HID * HID; idx += 256) w[idx] = W[idx];
    __syncthreads();

    const int wave = threadIdx.x >> 5;
    const int lane = threadIdx.x & 31;
    const int row0 = blockIdx.x * 32 + (wave >> 2) * 16;
    const int col0 = (wave & 3) * 16;

    // 32-bit A 16x4 layout: lanes 0-15 -> K = k0,k0+1 ; lanes 16-31 -> K = k0+2,k0+3
    const int m  = lane & 15;            // M index (A) / N index (B)
    const int kk = (lane >> 4) << 1;     // K sub-offset for this lane half

    int arow = row0 + m;                 // clamp for tail (N % 32 == 0 in practice)
    if (arow > N - 1) arow = N - 1;
    const float* ap = A + (size_t)arow * HID + kk;
    const float* bp = w + kk * HID + col0 + m;

    v8f c = {};
#pragma unroll
    for (int k0 = 0; k0 < HID; k0 += 4) {
        v2f a, b;
        a.x = ap[k0];
        a.y = ap[k0 + 1];
        b.x = bp[k0 * HID];              // B[k0+kk  ][col0+n]
        b.y = bp[k0 * HID + HID];        // B[k0+kk+1][col0+n]
        c = __builtin_amdgcn_wmma_f32_16x16x4_f32(
                /*neg_a=*/false, a, /*neg_b=*/false, b,
                /*c_mod=*/(short)0, c, /*reuse_a=*/false, /*reuse_b=*/false);
    }

    // C/D layout: VGPR r holds row M=r (lanes 0-15) / M=8+r (lanes 16-31)
    const int mbase = row0 + ((lane >> 4) << 3);
#pragma unroll
    for (int r = 0; r < 8; ++r) {
        int rr = mbase + r;
        if (rr < N) Bout[(size_t)rr * HID + col0 + m] = c[r];
    }
}

// ---------------- self-loop init of accumulator: agg = dinv^2 * hlin ----------------
__global__ void k_selfloop(const float* __restrict__ hlin, const float* __restrict__ dinv,
                           float* __restrict__ agg, int N) {
    int tid = blockIdx.x * blockDim.x + threadIdx.x;
    if (tid >= N * HID) return;
    float d = dinv[tid >> 6];
    agg[tid] = d * d * hlin[tid];
}

// ---------------- edge scatter: 64 threads/edge, coalesced gather + f32 atomics ----------------
__global__ void k_edge(const int* __restrict__ src, const int* __restrict__ dst,
                       const float* __restrict__ dinv, const float* __restrict__ hlin,
                       float* __restrict__ agg, int E) {
    int t = blockIdx.x * blockDim.x + threadIdx.x;
    int e = t >> 6;
    if (e >= E) return;
    int f = t & 63;
    int s = src[e], d = dst[e];
    float nrm = dinv[s] * dinv[d];
    atomicAdd(&agg[(size_t)d * HID + f], hlin[(size_t)s * HID + f] * nrm);
}

// ---------------- bias + relu ----------------
__global__ void k_finalize(const float* __restrict__ agg, const float* __restrict__ b,
                           float* __restrict__ h, int N) {
    int tid = blockIdx.x * blockDim.x + threadIdx.x;
    if (tid >= N * HID) return;
    float v = agg[tid] + b[tid & 63];
    h[tid] = v > 0.0f ? v : 0.0f;
}

// ---------------- pooling ----------------
__global__ void k_pool_zero(float* __restrict__ psum, int* __restrict__ pmax,
                            float* __restrict__ pcnt, int G) {
    int tid = blockIdx.x * blockDim.x + threadIdx.x;
    if (tid < G * HID) { psum[tid] = 0.0f; pmax[tid] = 0; }  // 0 bits == 0.0f
    if (tid < G) pcnt[tid] = 0.0f;
}

__global__ void k_pool(const float* __restrict__ h, const int* __restrict__ batch,
                       float* __restrict__ psum, int* __restrict__ pmax,
                       float* __restrict__ pcnt, int N) {
    int t = blockIdx.x * blockDim.x + threadIdx.x;
    int i = t >> 6;
    if (i >= N) return;
    int f = t & 63;
    int g = batch[i];
    float v = h[(size_t)i * HID + f];
    atomicAdd(&psum[g * HID + f], v);
    // post-ReLU v >= 0: int compare on bit pattern is monotone
    atomicMax(&pmax[g * HID + f], __float_as_int(v));
    if (f == 0) atomicAdd(&pcnt[g], 1.0f);
}

// ---------------- head: concat([mean,max]) @ Wl + bl ----------------
__global__ void k_head(const float* __restrict__ psum, const int* __restrict__ pmax,
                       const float* __restrict__ pcnt, const float* __restrict__ Wl,
                       const float* __restrict__ bl, float* __restrict__ out, int G) {
    int t = blockIdx.x * blockDim.x + threadIdx.x;
    if (t >= G * HID) return;
    int g = t >> 6, o = t & 63;
    float cnt = pcnt[g];
    float inv = 1.0f / (cnt > 1.0f ? cnt : 1.0f);
    const float* ms = psum + g * HID;
    const int*   mx = pmax + g * HID;
    float acc = bl[o];
#pragma unroll 4
    for (int k = 0; k < HID; ++k)
        acc = fmaf(ms[k] * inv, Wl[k * HID + o], acc);
#pragma unroll 4
    for (int k = 0; k < HID; ++k)
        acc = fmaf(__int_as_float(mx[k]), Wl[(HID + k) * HID + o], acc);
    out[t] = acc;
}

extern "C" void kernel_launch(void* const* d_in, const int* in_sizes, int n_in,
                              void* d_out, int out_size, void* d_ws, size_t ws_size,
                              hipStream_t stream) {
    const float* x     = (const float*)d_in[0];
    const int*   ei    = (const int*)  d_in[1];
    const int*   batch = (const int*)  d_in[3];
    const float* W1 = (const float*)d_in[4];  const float* b1 = (const float*)d_in[5];
    const float* W2 = (const float*)d_in[6];  const float* b2 = (const float*)d_in[7];
    const float* W3 = (const float*)d_in[8];  const float* b3 = (const float*)d_in[9];
    const float* Wl = (const float*)d_in[10]; const float* bl = (const float*)d_in[11];

    const int N = in_sizes[0] / 9;
    const int E = in_sizes[1] / 2;
    const int G = out_size / HID;
    const int* src = ei;
    const int* dst = ei + E;

    // ---- carve workspace (256B-aligned slabs) ----
    char* ws = (char*)d_ws;
    size_t off = 0;
    auto carve = [&](size_t bytes) -> void* {
        void* p = ws + off;
        off += (bytes + 255) & ~(size_t)255;
        return p;
    };
    float* dinv = (float*)carve((size_t)N * 4);
    float* hA   = (float*)carve((size_t)N * HID * 4);   // activations (GEMM input)
    float* hB   = (float*)carve((size_t)N * HID * 4);   // linear output
    float* hC   = (float*)carve((size_t)N * HID * 4);   // aggregation accumulator
    float* psum = (float*)carve((size_t)G * HID * 4);
    int*   pmax = (int*)  carve((size_t)G * HID * 4);
    float* pcnt = (float*)carve((size_t)G * 4);

    const int B = 256;
    const int gN   = (N + B - 1) / B;
    const int gE   = (E + B - 1) / B;
    const int gNF  = (N * HID + B - 1) / B;
    const int gEF  = (E * HID + B - 1) / B;
    const int gGF  = (G * HID + B - 1) / B;
    const int gMMA = (N + 31) / 32;

    // ---- degree normalization (graph-invariant across layers) ----
    k_deg_init <<<gN, B, 0, stream>>>(dinv, N);
    k_deg_count<<<gE, B, 0, stream>>>(dst, dinv, E);
    k_deg_rsqrt<<<gN, B, 0, stream>>>(dinv, N);
    k_pool_zero<<<gGF, B, 0, stream>>>(psum, pmax, pcnt, G);

    // ---- layer 1 ----
    k_lin_in  <<<gNF, B, 0, stream>>>(x, W1, hB, N);
    k_selfloop<<<gNF, B, 0, stream>>>(hB, dinv, hC, N);
    k_edge    <<<gEF, B, 0, stream>>>(src, dst, dinv, hB, hC, E);
    k_finalize<<<gNF, B, 0, stream>>>(hC, b1, hA, N);

    // ---- layer 2 (WMMA) ----
    k_gemm_wmma<<<gMMA, B, 0, stream>>>(hA, W2, hB, N);
    k_selfloop <<<gNF, B, 0, stream>>>(hB, dinv, hC, N);
    k_edge     <<<gEF, B, 0, stream>>>(src, dst, dinv, hB, hC, E);
    k_finalize <<<gNF, B, 0, stream>>>(hC, b2, hA, N);

    // ---- layer 3 (WMMA) ----
    k_gemm_wmma<<<gMMA, B, 0, stream>>>(hA, W3, hB, N);
    k_selfloop <<<gNF, B, 0, stream>>>(hB, dinv, hC, N);
    k_edge     <<<gEF, B, 0, stream>>>(src, dst, dinv, hB, hC, E);
    k_finalize <<<gNF, B, 0, stream>>>(hC, b3, hA, N);

    // ---- pooling + head ----
    k_pool<<<gNF, B, 0, stream>>>(hA, batch, psum, pmax, pcnt, N);
    k_head<<<gGF, B, 0, stream>>>(psum, pmax, pcnt, Wl, bl, (float*)d_out, G);
}